// HeteroGNN_84559316123741
// MI455X (gfx1250) — compile-verified
//
#include <hip/hip_runtime.h>
#include <hip/hip_bf16.h>

#define NP_BATCH 100000
#define NT_BATCH 200000
#define NN       (NP_BATCH + NT_BATCH)
#define E_EDGES  2000000
#define NPRED    100000
#define DD       64

typedef __attribute__((ext_vector_type(2))) float v2f;
typedef __attribute__((ext_vector_type(8))) float v8f;

// fp32 WMMA: D(16x16) = A(16x4) * B(4x16) + C, full wave32 op
__device__ __forceinline__ v8f wmma_f32(v2f a, v2f b, v8f c) {
    return __builtin_amdgcn_wmma_f32_16x16x4_f32(
        /*neg_a=*/false, a, /*neg_b=*/false, b,
        /*c_mod=*/(short)0, c, /*reuse_a=*/false, /*reuse_b=*/false);
}

// ---------------------------------------------------------------------------
// 1) Embedding gather: x[i] = emb[type(i)][id[i]]; also seed JK sum out = x
// ---------------------------------------------------------------------------
__global__ void __launch_bounds__(256)
gather_kernel(const float* __restrict__ embP, const float* __restrict__ embT,
              const int* __restrict__ idP, const int* __restrict__ idT,
              float* __restrict__ x, float* __restrict__ out_sum) {
    long tid  = (long)blockIdx.x * 256 + threadIdx.x;   // one float4 per thread
    long node = tid >> 4;
    int  ch   = (int)(tid & 15);
    if (node >= NN) return;
    const float* srcrow;
    if (node < NP_BATCH) srcrow = embP + (long)idP[node] * DD;
    else                 srcrow = embT + (long)idT[node - NP_BATCH] * DD;
    float4 v = *(const float4*)(srcrow + ch * 4);
    *(float4*)(x + node * DD + ch * 4)       = v;
    *(float4*)(out_sum + node * DD + ch * 4) = v;
}

// ---------------------------------------------------------------------------
// 2) Edge scatter: agg[dst] += x[src] (fp32 atomics), cnt[dst] += 1
//    32 lanes per edge, 2 floats per lane. Both relations in one launch;
//    destination sets are disjoint so one agg/cnt buffer is correct.
// ---------------------------------------------------------------------------
__global__ void __launch_bounds__(256)
scatter_kernel(const float* __restrict__ x,
               const int* __restrict__ e_pt, const int* __restrict__ e_tp,
               float* __restrict__ agg, float* __restrict__ cnt) {
    long tid  = (long)blockIdx.x * 256 + threadIdx.x;
    long gid  = tid >> 5;           // edge slot over both relations
    int  lane = (int)(tid & 31);
    if (gid >= 2L * E_EDGES) return;
    int src, dst;
    if (gid < E_EDGES) {            // playlist -> track (W[0] side)
        int e = (int)gid;
        src = e_pt[e];
        dst = e_pt[E_EDGES + e] + NP_BATCH;
    } else {                        // track -> playlist (W[1] side)
        int e = (int)(gid - E_EDGES);
        src = e_tp[e] + NP_BATCH;
        dst = e_tp[E_EDGES + e];
    }
    const float2 v = *(const float2*)(x + (long)src * DD + lane * 2);
    float* dp = agg + (long)dst * DD + lane * 2;
    atomicAdd(dp,     v.x);
    atomicAdd(dp + 1, v.y);
    if (lane == 0) atomicAdd(cnt + dst, 1.0f);
}

// ---------------------------------------------------------------------------
// 3) Node update via fp32 WMMA:
//    h = relu(x@root + b + (agg/cnt)@W_sel); x <- h; out_sum += h
//    One wave per 16-node tile; weights staged in LDS (row stride 72 floats
//    => the two half-wave K rows map to disjoint bank sets).
// ---------------------------------------------------------------------------
#define LSTR 72
__global__ void __launch_bounds__(256)
node_update_kernel(float* __restrict__ x,
                   const float* __restrict__ agg, const float* __restrict__ cnt,
                   const float* __restrict__ Wrel, const float* __restrict__ root,
                   const float* __restrict__ bias, float* __restrict__ out_sum) {
    __shared__ float lds[3 * DD * LSTR + DD];
    float* Lroot = lds;
    float* LW0   = lds + DD * LSTR;
    float* LW1   = lds + 2 * DD * LSTR;
    float* Lb    = lds + 3 * DD * LSTR;
    for (int i = threadIdx.x; i < DD * DD; i += 256) {
        int k = i >> 6, c = i & 63;
        Lroot[k * LSTR + c] = root[i];
        LW0[k * LSTR + c]   = Wrel[i];               // relation 0 (dst=track)
        LW1[k * LSTR + c]   = Wrel[DD * DD + i];     // relation 1 (dst=playlist)
    }
    if (threadIdx.x < DD) Lb[threadIdx.x] = bias[threadIdx.x];
    __syncthreads();

    int tile     = blockIdx.x * 8 + (threadIdx.x >> 5);
    int tilebase = tile * 16;
    if (tilebase >= NN) return;                      // wave-uniform exit

    int lane  = threadIdx.x & 31;
    int mrow  = lane & 15;
    int khalf = lane >> 4;
    long row  = tilebase + mrow;
    const float* xrow = x   + row * DD;
    const float* arow = agg + row * DD;
    float c   = cnt[row];
    float inv = (c > 0.f) ? (1.f / c) : 0.f;
    const float* LW = (tilebase < NP_BATCH) ? LW1 : LW0;

    v8f acc[4];
#pragma unroll
    for (int n = 0; n < 4; ++n) {
        float bv = Lb[n * 16 + mrow];
        v8f t = {bv, bv, bv, bv, bv, bv, bv, bv};
        acc[n] = t;
    }
#pragma unroll
    for (int ks = 0; ks < 16; ++ks) {
        int k0 = ks * 4 + khalf * 2;
        float2 av = *(const float2*)(xrow + k0);
        float2 gv = *(const float2*)(arow + k0);
        v2f a = {av.x, av.y};
        v2f m = {gv.x * inv, gv.y * inv};
#pragma unroll
        for (int n = 0; n < 4; ++n) {
            int cb = n * 16 + mrow;
            v2f br = {Lroot[k0 * LSTR + cb], Lroot[(k0 + 1) * LSTR + cb]};
            v2f bw = {LW[k0 * LSTR + cb],    LW[(k0 + 1) * LSTR + cb]};
            acc[n] = wmma_f32(a, br, acc[n]);
            acc[n] = wmma_f32(m, bw, acc[n]);
        }
    }
    // C/D layout: vgpr v -> row (v + 8*khalf), col (n*16 + mrow); each (row,col)
    // is owned by exactly one lane -> plain RMW on out_sum is race-free.
    long rbase = tilebase + khalf * 8;
#pragma unroll
    for (int n = 0; n < 4; ++n) {
#pragma unroll
        for (int v = 0; v < 8; ++v) {
            float h = acc[n][v];
            h = (h > 0.f) ? h : 0.f;
            long off = (rbase + v) * DD + n * 16 + mrow;
            x[off] = h;
            out_sum[off] += h;
        }
    }
}

// ---------------------------------------------------------------------------
// 4) Link prediction: dot(out[p], out[NP_BATCH+t]) * (1/3)*(1/3)
// ---------------------------------------------------------------------------
__global__ void __launch_bounds__(256)
predict_kernel(const float* __restrict__ out_sum, const int* __restrict__ eli,
               float* __restrict__ pred) {
    long tid = (long)blockIdx.x * 256 + threadIdx.x;
    long p   = tid >> 4;
    int  sub = (int)(tid & 15);
    if (p >= NPRED) return;
    long pr = eli[p];
    long tr = (long)eli[NPRED + p] + NP_BATCH;
    float4 a = *(const float4*)(out_sum + pr * DD + sub * 4);
    float4 b = *(const float4*)(out_sum + tr * DD + sub * 4);
    float d = a.x * b.x + a.y * b.y + a.z * b.z + a.w * b.w;
    d += __shfl_xor(d, 1);
    d += __shfl_xor(d, 2);
    d += __shfl_xor(d, 4);
    d += __shfl_xor(d, 8);
    if (sub == 0) pred[p] = d * (1.0f / 9.0f);
}

extern "C" void kernel_launch(void* const* d_in, const int* in_sizes, int n_in,
                              void* d_out, int out_size, void* d_ws, size_t ws_size,
                              hipStream_t stream) {
    const float* embP  = (const float*)d_in[0];
    const float* embT  = (const float*)d_in[1];
    const float* W1    = (const float*)d_in[2];
    const float* root1 = (const float*)d_in[3];
    const float* b1    = (const float*)d_in[4];
    const float* W2    = (const float*)d_in[5];
    const float* root2 = (const float*)d_in[6];
    const float* b2    = (const float*)d_in[7];
    const int*   idP   = (const int*)d_in[8];
    const int*   idT   = (const int*)d_in[9];
    const int*   ept   = (const int*)d_in[10];
    const int*   etp   = (const int*)d_in[11];
    const int*   eli   = (const int*)d_in[12];
    float* pred = (float*)d_out;

    float* ws      = (float*)d_ws;
    float* x       = ws;                              // [NN, 64]
    float* out_sum = x + (size_t)NN * DD;             // [NN, 64]
    float* agg     = out_sum + (size_t)NN * DD;       // [NN, 64]
    float* cnt     = agg + (size_t)NN * DD;           // [NN]   (contiguous w/ agg)

    long gthreads = (long)NN * 16;
    gather_kernel<<<(int)((gthreads + 255) / 256), 256, 0, stream>>>(
        embP, embT, idP, idT, x, out_sum);

    const float* Ws[2] = {W1, W2};
    const float* Rs[2] = {root1, root2};
    const float* Bs[2] = {b1, b2};
    long sthreads = 2L * E_EDGES * 32;
    int  tiles    = NN / 16;                          // 18750, exact
    for (int l = 0; l < 2; ++l) {
        hipMemsetAsync(agg, 0, ((size_t)NN * DD + NN) * sizeof(float), stream);
        scatter_kernel<<<(int)((sthreads + 255) / 256), 256, 0, stream>>>(
            x, ept, etp, agg, cnt);
        node_update_kernel<<<(tiles + 7) / 8, 256, 0, stream>>>(
            x, agg, cnt, Ws[l], Rs[l], Bs[l], out_sum);
    }

    long pthreads = (long)NPRED * 16;
    predict_kernel<<<(int)((pthreads + 255) / 256), 256, 0, stream>>>(
        out_sum, eli, pred);
}